// Model_39676907886106
// MI455X (gfx1250) — compile-verified
//
#include <hip/hip_runtime.h>

typedef float v2f __attribute__((ext_vector_type(2)));
typedef float v8f __attribute__((ext_vector_type(8)));
typedef unsigned int u32x4 __attribute__((ext_vector_type(4)));
typedef int          i32x8 __attribute__((ext_vector_type(8)));

#define NN    384
#define TILE  1024   // 32*32 elements per matrix tile
#define WAVES 4

// ---- Tensor Data Mover helpers (CDNA5 TDM, CDNA5 ISA) ----
// Inline asm (not the clang builtin) so the code is portable across the
// 5-arg (ROCm 7.2) and 6-arg (amdgpu-toolchain) builtin variants.
// Groups 2/3 omitted -> tensor up to 2D.

__device__ __forceinline__ void tdm_load_to_lds(u32x4 g0, i32x8 g1) {
    asm volatile("tensor_load_to_lds %0, %1" :: "s"(g0), "s"(g1) : "memory");
}
__device__ __forceinline__ void tdm_store_from_lds(u32x4 g0, i32x8 g1) {
    asm volatile("tensor_store_from_lds %0, %1" :: "s"(g0), "s"(g1) : "memory");
}

// D# group 0: count=1 (valid user descriptor), lds_addr, 57-bit global addr,
// type=2 ("image") in bits [127:126].
__device__ __forceinline__ u32x4 tdm_g0(unsigned int lds_addr,
                                        unsigned long long gaddr) {
    u32x4 g0;
    g0.x = 1u;                                     // count=1, is_restore=0
    g0.y = lds_addr;                               // LDS byte address
    g0.z = (unsigned int)gaddr;                    // global_addr[31:0]
    g0.w = (unsigned int)(gaddr >> 32) | (2u << 30); // global_addr[56:32] | type=2
    return g0;
}

// D# group 1 describing one contiguous 16 KB row: data_size=4B,
// tensor_dim0 = tile_dim0 = 4096 elements, tensor_dim1 = tile_dim1 = 1.
__device__ __forceinline__ i32x8 tdm_g1_16kb() {
    i32x8 g1;
    g1[0] = 0x00020000;            // [15:0] wg_mask=0, [17:16] data_size=2 (4B)
    g1[1] = (4096 & 0xFFFF) << 16; // [47:32] atomic_barrier=0, [63:48] dim0 lo16
    g1[2] = (1 << 16);             // [79:64] dim0 hi16=0, [95:80] dim1 lo16=1
    g1[3] = (4096 << 16);          // [111:96] dim1 hi16=0, [127:112] tile_dim0=4096
    g1[4] = 1;                     // [143:128] tile_dim1=1, [159:144] tile_dim2=0
    g1[5] = 4096;                  // [191:160] tensor_dim0_stride lo32
    g1[6] = 0;                     // stride hi / dim1_stride lo
    g1[7] = 0;
    return g1;
}

__global__ __launch_bounds__(128, 1)
void attn_softmax_drop_kernel(const float* __restrict__ x1,
                              const float* __restrict__ mask,
                              float* __restrict__ out) {
    __shared__ float ldsKa[TILE];            // 4 KB : x1[a], shared by 4 waves
    __shared__ float ldsQV[WAVES][TILE];     // 16 KB: x1[b] per wave (Q == V)
    __shared__ float ldsMP[WAVES][TILE];     // 16 KB: mask -> P -> out (in place)

    const int a    = blockIdx.y;
    const int wave = threadIdx.x >> 5;
    const int lane = threadIdx.x & 31;
    const int b    = blockIdx.x * WAVES + wave;
    const int half = lane >> 4;
    const int l16  = lane & 15;

    // Block-contiguous 16 KB regions: the 4 waves' mask/out tiles for
    // b = blockIdx.x*4 .. +3 are adjacent in memory (b varies fastest).
    const size_t blk_off = ((size_t)a * NN + (size_t)blockIdx.x * WAVES) * TILE;

    // ---- stage x1[a] / x1[b] into LDS (tiny, L2-resident) ----
    {
        const float4* src = (const float4*)(x1 + a * TILE);
        float4*       dst = (float4*)ldsKa;
        #pragma unroll
        for (int i = threadIdx.x; i < TILE / 4; i += 128) dst[i] = src[i];
    }
    {
        const float4* src = (const float4*)(x1 + b * TILE);
        float4*       dst = (float4*)ldsQV[wave];
        #pragma unroll
        for (int i = lane; i < TILE / 4; i += 32) dst[i] = src[i];
    }

    // ---- kick the TDM: DMA the block's 16 KB of dropout mask into ldsMP.
    //      Runs underneath matmul 1; costs zero VGPRs and zero VMEM issue
    //      slots in the worker waves. Descriptor is block-uniform -> SGPRs.
    if (wave == 0) {
        tdm_load_to_lds(
            tdm_g0((unsigned int)(uintptr_t)(void*)ldsMP,
                   (unsigned long long)(uintptr_t)(const void*)(mask + blk_off)),
            tdm_g1_16kb());
    }
    __syncthreads();   // x1 staging visible to all waves

    // ---- matmul 1: qk = Q @ Ka^T  (Q = x1[b], Ka = x1[a]) ----
    const float* Q = ldsQV[wave];
    v8f acc[2][2];
    #pragma unroll
    for (int mt = 0; mt < 2; ++mt) {
        #pragma unroll
        for (int nt = 0; nt < 2; ++nt) {
            v8f c = {};
            #pragma unroll
            for (int k8 = 0; k8 < 8; ++k8) {
                const int k0 = k8 * 4 + 2 * half;
                v2f av = *(const v2f*)&Q[(mt * 16 + l16) * 32 + k0];
                v2f bv = *(const v2f*)&ldsKa[(nt * 16 + l16) * 32 + k0];
                c = __builtin_amdgcn_wmma_f32_16x16x4_f32(
                        false, av, false, bv, (short)0, c, false, false);
            }
            acc[mt][nt] = c;
        }
    }

    // ---- TDM completion fence: wave 0 owns TENSORcnt; barrier releases all ----
    if (wave == 0) __builtin_amdgcn_s_wait_tensorcnt(0);
    __syncthreads();

    // ---- softmax (scale = sqrt(1) = 1) in registers + dropout, P in place ----
    float* MP = ldsMP[wave];
    #pragma unroll
    for (int mt = 0; mt < 2; ++mt) {
        #pragma unroll
        for (int r = 0; r < 8; ++r) {
            float q0 = acc[mt][0][r];
            float q1 = acc[mt][1][r];
            float mx = fmaxf(q0, q1);
            #pragma unroll
            for (int s = 8; s >= 1; s >>= 1) mx = fmaxf(mx, __shfl_xor(mx, s, 32));
            float e0  = __expf(q0 - mx);
            float e1  = __expf(q1 - mx);
            float sum = e0 + e1;
            #pragma unroll
            for (int s = 8; s >= 1; s >>= 1) sum += __shfl_xor(sum, s, 32);
            const float inv = 1.0f / sum;
            const int   m   = mt * 16 + r + 8 * half;
            float p0 = e0 * inv * MP[m * 32 + l16];        // same lane, same addr:
            float p1 = e1 * inv * MP[m * 32 + 16 + l16];   // read-then-write safe
            MP[m * 32 + l16]      = p0;
            MP[m * 32 + 16 + l16] = p1;
        }
    }
    __syncthreads();

    // ---- matmul 2: out = P @ V  (V = x1[b], non-transposed B operand) ----
    const float* V = ldsQV[wave];
    v8f oacc[2][2];
    #pragma unroll
    for (int mt = 0; mt < 2; ++mt) {
        #pragma unroll
        for (int kt = 0; kt < 2; ++kt) {
            v8f c = {};
            #pragma unroll
            for (int k8 = 0; k8 < 8; ++k8) {
                const int k0 = k8 * 4 + 2 * half;
                v2f av = *(const v2f*)&MP[(mt * 16 + l16) * 32 + k0];
                v2f bv;
                bv.x = V[(k0    ) * 32 + kt * 16 + l16];
                bv.y = V[(k0 + 1) * 32 + kt * 16 + l16];
                c = __builtin_amdgcn_wmma_f32_16x16x4_f32(
                        false, av, false, bv, (short)0, c, false, false);
            }
            oacc[mt][kt] = c;
        }
    }
    __syncthreads();   // all P reads done before overwriting ldsMP with output

    // ---- deposit output tile into LDS (row-major) ----
    #pragma unroll
    for (int mt = 0; mt < 2; ++mt) {
        #pragma unroll
        for (int r = 0; r < 8; ++r) {
            const int m = mt * 16 + r + 8 * half;
            MP[m * 32 + l16]      = oacc[mt][0][r];
            MP[m * 32 + 16 + l16] = oacc[mt][1][r];
        }
    }
    __syncthreads();

    // ---- TDM store: one contiguous 16 KB burst LDS -> HBM for the block ----
    if (wave == 0) {
        tdm_store_from_lds(
            tdm_g0((unsigned int)(uintptr_t)(void*)ldsMP,
                   (unsigned long long)(uintptr_t)(void*)(out + blk_off)),
            tdm_g1_16kb());
        __builtin_amdgcn_s_wait_tensorcnt(0);
    }
}

extern "C" void kernel_launch(void* const* d_in, const int* in_sizes, int n_in,
                              void* d_out, int out_size, void* d_ws, size_t ws_size,
                              hipStream_t stream) {
    const float* x1   = (const float*)d_in[0];   // (1, 384, 32, 32) f32
    const float* mask = (const float*)d_in[1];   // (384, 384, 32, 32) f32
    float*       out  = (float*)d_out;           // (384, 384, 32, 32) f32

    dim3 grid(NN / WAVES, NN);   // (96, 384): x = b-group of 4, y = a
    dim3 block(128);             // 4 wave32
    attn_softmax_drop_kernel<<<grid, block, 0, stream>>>(x1, mask, out);
}